// TopKMemoryRetriever_59382217834561
// MI455X (gfx1250) — compile-verified
//
#include <hip/hip_runtime.h>

// CDNA5 / gfx1250. Wave32. f32 WMMA 16x16x4.
typedef __attribute__((ext_vector_type(2))) float v2f;
typedef __attribute__((ext_vector_type(8))) float v8f;

#define HD     256
#define NKEYS  32768
#define NB     32
#define TOPK   16
#define SCALE  0.0625f   // 1/sqrt(256)

// ---------------------------------------------------------------------------
// Stage 1 (tiny): v[b,:] = Wk^T * (Wq * q[b,:])   -- one 1024-thread WG,
// pq staged in LDS, both GEMMs via V_WMMA_F32_16X16X4_F32.
// f32 WMMA lane layout (wave32): lane = 16*h + m (h=0/1, m=0..15).
//   A (16x4): lane holds A[m][2h], A[m][2h+1]
//   B (4x16): lane holds B[2h][m], B[2h+1][m]
//   D (16x16): VGPR r on lane -> D[r + 8h][m]
// ---------------------------------------------------------------------------
__global__ __launch_bounds__(1024) void proj_kernel(
    const float* __restrict__ q,    // [NB, HD]
    const float* __restrict__ Wq,   // [HD, HD] (torch Linear: out x in)
    const float* __restrict__ Wk,   // [HD, HD]
    float* __restrict__ v_out)      // [NB, HD]
{
  __shared__ float pq[NB * HD];
  const int t    = threadIdx.x;
  const int wave = t >> 5;
  const int lane = t & 31;
  const int hh   = lane >> 4;
  const int m    = lane & 15;

  // ---- phase 1: pq[b,h] = sum_d q[b,d] * Wq[h,d] ----
  {
    const int bt = wave >> 4;   // 0..1  (32 batches / 16)
    const int ht = wave & 15;   // 0..15 (256 hidden / 16)
    v8f acc = {};
    const float* arow = q  + (bt * 16 + m) * HD;   // A[m][k] = q[b, k]
    const float* brow = Wq + (ht * 16 + m) * HD;   // B[k][n] = Wq[h, k] (n=m lane slot)
    for (int i = 0; i < HD / 4; ++i) {
      const int c = i * 4 + 2 * hh;
      v2f a;  a.x  = arow[c]; a.y  = arow[c + 1];
      v2f bm; bm.x = brow[c]; bm.y = brow[c + 1];
      acc = __builtin_amdgcn_wmma_f32_16x16x4_f32(false, a, false, bm,
                                                  (short)0, acc, false, false);
    }
    for (int r = 0; r < 8; ++r)
      pq[(bt * 16 + r + 8 * hh) * HD + ht * 16 + m] = acc[r];
  }
  __syncthreads();

  // ---- phase 2: v[b,d] = sum_h pq[b,h] * Wk[h,d] ----
  {
    const int bt = wave >> 4;
    const int dt = wave & 15;
    v8f acc = {};
    const float* arow = pq + (bt * 16 + m) * HD;   // A[m][k] = pq[b, k]
    for (int i = 0; i < HD / 4; ++i) {
      const int c = i * 4 + 2 * hh;
      v2f a;  a.x  = arow[c]; a.y  = arow[c + 1];
      v2f bm;                                      // B[k][n] = Wk[k, d]
      bm.x = Wk[(c    ) * HD + dt * 16 + m];
      bm.y = Wk[(c + 1) * HD + dt * 16 + m];
      acc = __builtin_amdgcn_wmma_f32_16x16x4_f32(false, a, false, bm,
                                                  (short)0, acc, false, false);
    }
    for (int r = 0; r < 8; ++r)
      v_out[(bt * 16 + r + 8 * hh) * HD + dt * 16 + m] = acc[r];
  }
}

// ---------------------------------------------------------------------------
// Stage 2 (HBM-bound stream): scores[b,n] = SCALE * keys[b,n,:] . v[b,:]
// One wave per 16-row key tile; K=256 accumulated over 64 chained WMMAs.
// B columns all replicate v, column N=0 of D is the result.
// Tiles entirely past segment_offset skip the key reads (halves HBM traffic).
// ---------------------------------------------------------------------------
__global__ __launch_bounds__(256) void scores_kernel(
    const float* __restrict__ keys,   // [NB, NKEYS, HD]
    const float* __restrict__ v,      // [NB, HD]
    const int*   __restrict__ seg,    // scalar
    float* __restrict__ scores)       // [NB, NKEYS]
{
  __shared__ float vsh[HD];
  const int t     = threadIdx.x;
  const int wave  = t >> 5;
  const int lane  = t & 31;
  const int hh    = lane >> 4;
  const int m     = lane & 15;
  const int b     = blockIdx.x >> 8;          // 256 blocks (2048 tiles) per batch
  const int local = blockIdx.x & 255;
  const int n0    = (local * 8 + wave) * 16;  // 8 waves/block, 16 rows/wave

  vsh[t] = v[b * HD + t];
  __syncthreads();

  const int  off    = seg[0];
  const bool maskOn = (off > 0) && (off < NKEYS);
  float* srow = scores + (size_t)b * NKEYS;

  if (maskOn && n0 >= off) {                  // fully-masked tile: no key reads
    if (lane < 16) srow[n0 + lane] = -__builtin_inff();
    return;
  }

  v8f acc = {};
  const float* arow = keys + ((size_t)b * NKEYS + n0 + m) * HD;
  for (int i = 0; i < HD / 4; ++i) {
    const int c = i * 4 + 2 * hh;
    v2f a;  a.x  = arow[c]; a.y  = arow[c + 1];
    v2f bm; bm.x = vsh[c];  bm.y = vsh[c + 1];
    acc = __builtin_amdgcn_wmma_f32_16x16x4_f32(false, a, false, bm,
                                                (short)0, acc, false, false);
  }
  if (m == 0) {                               // lanes 0 & 16 hold column N=0
    const int base = n0 + 8 * hh;
    for (int r = 0; r < 8; ++r) {
      const int n = base + r;
      float val = acc[r] * SCALE;
      if (maskOn && n >= off) val = -__builtin_inff();
      srow[n] = val;
    }
  }
}

// ---------------------------------------------------------------------------
// Stage 3: per-batch top-16 + softmax. Whole 128KB score row lives in LDS
// (CDNA5: 320KB per WGP). 16 rounds of block argmax with -inf knockout;
// ties resolve to the lowest index (matches jax.lax.top_k).
// ---------------------------------------------------------------------------
__global__ __launch_bounds__(256) void topk_kernel(
    const float* __restrict__ scores,  // [NB, NKEYS]
    float* __restrict__ out_w,         // [NB, TOPK]
    float* __restrict__ out_i)         // [NB, TOPK] (indices as float)
{
  extern __shared__ float srow[];      // NKEYS floats (128 KB dynamic LDS)
  __shared__ float rv[256];
  __shared__ int   ri[256];
  __shared__ float topv[TOPK];
  __shared__ int   topi[TOPK];

  const int b = blockIdx.x;
  const int t = threadIdx.x;

  for (int n = t; n < NKEYS; n += 256)
    srow[n] = scores[(size_t)b * NKEYS + n];
  __syncthreads();

  for (int k = 0; k < TOPK; ++k) {
    float bv = -__builtin_inff();
    int   bi = 0x7fffffff;
    for (int n = t; n < NKEYS; n += 256) {
      const float x = srow[n];
      if (x > bv || (x == bv && n < bi)) { bv = x; bi = n; }
    }
    rv[t] = bv; ri[t] = bi;
    __syncthreads();
    for (int s = 128; s > 0; s >>= 1) {
      if (t < s) {
        const float xv = rv[t + s]; const int xi = ri[t + s];
        if (xv > rv[t] || (xv == rv[t] && xi < ri[t])) { rv[t] = xv; ri[t] = xi; }
      }
      __syncthreads();
    }
    if (t == 0) {
      topv[k] = rv[0];
      topi[k] = ri[0];
      srow[ri[0]] = -__builtin_inff();
    }
    __syncthreads();
  }

  if (t == 0) {
    const float mx = topv[0];
    float e[TOPK], s = 0.0f;
    for (int k = 0; k < TOPK; ++k) { e[k] = __expf(topv[k] - mx); s += e[k]; }
    const float inv = 1.0f / s;
    for (int k = 0; k < TOPK; ++k) {
      out_w[b * TOPK + k] = e[k] * inv;
      out_i[b * TOPK + k] = (float)topi[k];
    }
  }
}

// ---------------------------------------------------------------------------
extern "C" void kernel_launch(void* const* d_in, const int* in_sizes, int n_in,
                              void* d_out, int out_size, void* d_ws, size_t ws_size,
                              hipStream_t stream) {
  const float* q    = (const float*)d_in[0];   // [32, 256]
  const float* keys = (const float*)d_in[1];   // [32, 32768, 256]
  const float* Wq   = (const float*)d_in[2];   // [256, 256]
  const float* Wk   = (const float*)d_in[3];   // [256, 256]
  const int*   seg  = (const int*)  d_in[4];   // scalar

  float* out   = (float*)d_out;
  float* out_w = out;                          // [32,16] routing weights
  float* out_i = out + NB * TOPK;              // [32,16] indices (as float)
  float* out_s = out + 2 * NB * TOPK;          // [32,32768] scores

  float* v_ws = (float*)d_ws;                  // [32,256] fused projection

  proj_kernel<<<1, 1024, 0, stream>>>(q, Wq, Wk, v_ws);

  const int nTiles = NB * (NKEYS / 16);        // 65536 tiles, 8 waves/block
  scores_kernel<<<nTiles / 8, 256, 0, stream>>>(keys, v_ws, seg, out_s);

  topk_kernel<<<NB, 256, NKEYS * sizeof(float), stream>>>(out_s, out_w, out_i);
}